// OlmoeMoe_44564580663483
// MI455X (gfx1250) — compile-verified
//
#include <hip/hip_runtime.h>
#include <hip/hip_bf16.h>

// Problem constants (from reference): B=1, N=2048, D=1024, I=512, E=64, K=8, S=1
#define T_TOK  2048
#define DMODEL 1024
#define IMLP   512
#define NEXP   64
#define TOPK   8
// LDS row strides (padded to dodge bank conflicts on row-strided A reads)
#define XS (DMODEL + 8)
#define HS (IMLP + 8)
// Weight K-panel staged per wave: 64 K-rows x 16 N-cols of f32 (4 KB)
#define KPANEL 64
#define PANEL_FLOATS (KPANEL * 16)        // 1024
#define WAVE_PANEL_FLOATS (4 * PANEL_FLOATS) // G0,G1,U0,U1 per wave

typedef __attribute__((ext_vector_type(2))) float v2f;
typedef __attribute__((ext_vector_type(8))) float v8f;

__device__ __forceinline__ v8f wmma_f32_k4(v2f a, v2f b, v8f c) {
  // V_WMMA_F32_16X16X4_F32: D = A(16x4 f32) x B(4x16 f32) + C(16x16 f32)
  return __builtin_amdgcn_wmma_f32_16x16x4_f32(
      false, a, false, b, (short)0, c, false, false);
}

__device__ __forceinline__ void atomic_add_f32(float* p, float v) {
  __hip_atomic_fetch_add(p, v, __ATOMIC_RELAXED, __HIP_MEMORY_SCOPE_AGENT);
}

// --- CDNA5 async global->LDS staging (ASYNCcnt-tracked fire-and-forget) ----
// Copy a 64x16 f32 weight tile (rows kp0..kp0+63, cols n0..n0+15 of a row-major
// [K x ldw] matrix) into a per-wave LDS panel. 8 x GLOBAL_LOAD_ASYNC_TO_LDS_B128,
// each lane moving 16B: lane -> (row = i*8 + lane/4, col4 = (lane%4)*4).
__device__ __forceinline__ void issue_panel_async(
    const float* __restrict__ W, int ldw, int kp0, int n0, int lane,
    float* panel) {
  const int r0 = lane >> 2;          // 0..7
  const int c4 = (lane & 3) << 2;    // 0,4,8,12
  const float* g = W + (size_t)(kp0 + r0) * ldw + n0 + c4;
  unsigned l = (unsigned)(uintptr_t)(panel + r0 * 16 + c4);
#pragma unroll
  for (int i = 0; i < 8; ++i) {
    asm volatile("global_load_async_to_lds_b128 %0, %1, off"
                 :: "v"(l), "v"(g) : "memory");
    g += (size_t)8 * ldw;
    l += 8 * 16 * sizeof(float);
  }
}

__device__ __forceinline__ void wait_async(void) {
  asm volatile("s_wait_asynccnt 0" ::: "memory");
}
__device__ __forceinline__ void wait_async_le8(void) {
  asm volatile("s_wait_asynccnt 8" ::: "memory");
}
__device__ __forceinline__ void wait_async_le16(void) {
  asm volatile("s_wait_asynccnt 16" ::: "memory");
}

// ---------------------------------------------------------------------------
// Router: logits = x @ gate_w^T, softmax(f32), top-8. One 64-thread block per
// token (thread j owns expert j).
// ---------------------------------------------------------------------------
__global__ __launch_bounds__(64) void router_kernel(
    const float* __restrict__ x, const float* __restrict__ gw,
    float* __restrict__ logits_out, int* __restrict__ ids_out,
    int* __restrict__ topk_id, float* __restrict__ topk_w) {
  const int t = blockIdx.x;
  const int j = threadIdx.x;
  const float* xr = x + (size_t)t * DMODEL;
  const float* gr = gw + (size_t)j * DMODEL;
  float acc = 0.f;
  for (int d = 0; d < DMODEL; d += 4) {
    float4 xv = *(const float4*)(xr + d);
    float4 gv = *(const float4*)(gr + d);
    acc += xv.x * gv.x + xv.y * gv.y + xv.z * gv.z + xv.w * gv.w;
  }
  logits_out[(size_t)t * NEXP + j] = acc;
  __shared__ float sl[NEXP];
  sl[j] = acc;
  __syncthreads();
  if (j == 0) {
    float mx = sl[0];
    for (int e = 1; e < NEXP; ++e) mx = fmaxf(mx, sl[e]);
    float p[NEXP];
    float sum = 0.f;
    for (int e = 0; e < NEXP; ++e) { p[e] = __expf(sl[e] - mx); sum += p[e]; }
    const float inv = 1.f / sum;
    for (int k = 0; k < TOPK; ++k) {
      int best = 0; float bv = -1.f;
      for (int e = 0; e < NEXP; ++e) {
        if (p[e] > bv) { bv = p[e]; best = e; }
      }
      p[best] = -2.f;
      topk_id[t * TOPK + k] = best;
      topk_w[t * TOPK + k] = bv * inv;
      ids_out[t * TOPK + k] = best;
    }
  }
}

// ---------------------------------------------------------------------------
// Deterministic dispatch (no atomics -> bitwise-stable bucket order).
// ---------------------------------------------------------------------------
__global__ __launch_bounds__(64) void count_kernel(
    const int* __restrict__ topk_id,
    unsigned* __restrict__ counts, unsigned* __restrict__ offsets) {
  __shared__ unsigned sc[NEXP];
  const int e = threadIdx.x;
  unsigned c = 0;
  for (int i = 0; i < T_TOK * TOPK; ++i) c += (topk_id[i] == e) ? 1u : 0u;
  sc[e] = c;
  counts[e] = c;
  __syncthreads();
  if (e == 0) {
    unsigned off = 0;
    for (int i = 0; i < NEXP; ++i) { offsets[i] = off; off += sc[i]; }
  }
}

__global__ __launch_bounds__(64) void scatter_kernel(
    const int* __restrict__ topk_id, const float* __restrict__ topk_w,
    const unsigned* __restrict__ offsets,
    int* __restrict__ bucket_tid, float* __restrict__ bucket_w) {
  const int e = threadIdx.x;
  unsigned slot = offsets[e];
  for (int i = 0; i < T_TOK * TOPK; ++i) {
    if (topk_id[i] == e) {
      bucket_tid[slot] = i >> 3;   // token id
      bucket_w[slot]   = topk_w[i];
      ++slot;
    }
  }
}

// ---------------------------------------------------------------------------
// Expert SwiGLU GEMM. One block = one expert x one 16-token tile. 4 wave32s.
// Weight B-operands are double-buffered 64x16 K-panels staged per wave via
// GLOBAL_LOAD_ASYNC_TO_LDS_B128; s_wait_asynccnt gates consumption so the
// next panel's HBM traffic overlaps the current panel's 16 WMMA K-steps.
// ---------------------------------------------------------------------------
__global__ __launch_bounds__(128) void expert_kernel(
    const float* __restrict__ x,
    const float* __restrict__ wg_all, const float* __restrict__ wu_all,
    const float* __restrict__ wd_all,
    float* __restrict__ y,
    const int* __restrict__ bucket_tid, const float* __restrict__ bucket_w,
    const unsigned* __restrict__ counts, const unsigned* __restrict__ offsets,
    int gathered) {
  extern __shared__ float smem[];
  float* Xs  = smem;                       // 16 x XS   activation panel
  float* Hs  = smem + 16 * XS;             // 16 x HS   swiglu intermediate
  float* Pan = smem + 16 * XS + 16 * HS;   // per-wave weight panels
  __shared__ int   tids[16];
  __shared__ float twts[16];

  int e, tile, cnt, base;
  if (gathered) {
    e = blockIdx.x >> 7;          // 128 max tiles per expert (2048/16)
    tile = blockIdx.x & 127;
    cnt = (int)counts[e];
    if (tile * 16 >= cnt) return; // uniform early-exit, whole block
    base = (int)offsets[e];
  } else {
    e = 0; tile = blockIdx.x; cnt = T_TOK; base = 0;
  }

  if (threadIdx.x < 16) {
    int row = tile * 16 + (int)threadIdx.x;
    if (gathered) {
      bool v = row < cnt;
      tids[threadIdx.x] = v ? bucket_tid[base + row] : 0;
      twts[threadIdx.x] = v ? bucket_w[base + row] : 0.f;  // pad rows add 0
    } else {
      tids[threadIdx.x] = row;
      twts[threadIdx.x] = 1.f;
    }
  }
  __syncthreads();

  // Gather the 16 x 1024 activation panel into LDS (float4 loads).
  for (int v = threadIdx.x; v < 16 * (DMODEL / 4); v += 128) {
    int row = v / (DMODEL / 4);
    int c4  = v % (DMODEL / 4);
    float4 val = *((const float4*)(x + (size_t)tids[row] * DMODEL) + c4);
    float* dst = Xs + row * XS + c4 * 4;
    dst[0] = val.x; dst[1] = val.y; dst[2] = val.z; dst[3] = val.w;
  }
  __syncthreads();

  const float* WG = wg_all + (size_t)e * DMODEL * IMLP;
  const float* WU = wu_all + (size_t)e * DMODEL * IMLP;
  const float* WD = wd_all + (size_t)e * IMLP * DMODEL;

  const int lane   = threadIdx.x & 31;
  const int wave   = threadIdx.x >> 5;
  const int m      = lane & 15;           // A row
  const int kh     = (lane >> 4) << 1;    // A/B K sub-offset: 0 or 2
  const int nL     = lane & 15;           // B/C column
  const int rowOff = (lane >> 4) << 3;    // C row offset: 0 or 8

  float* waveP = Pan + wave * WAVE_PANEL_FLOATS;
  float* Pg0 = waveP;                     // G panel double-buffer
  float* Pg1 = waveP + PANEL_FLOATS;
  float* Pu0 = waveP + 2 * PANEL_FLOATS;  // U panel double-buffer
  float* Pu1 = waveP + 3 * PANEL_FLOATS;

  // ---- Phase A: gate & up projections + SwiGLU ----
  for (int nt = wave; nt < IMLP / 16; nt += 4) {
    const int n0 = nt * 16;
    v8f cg = {}; v8f cu = {};
    issue_panel_async(WG, IMLP, 0, n0, lane, Pg0);
    issue_panel_async(WU, IMLP, 0, n0, lane, Pu0);
    for (int p = 0; p < DMODEL / KPANEL; ++p) {
      float* pg = (p & 1) ? Pg1 : Pg0;
      float* pu = (p & 1) ? Pu1 : Pu0;
      if (p + 1 < DMODEL / KPANEL) {
        issue_panel_async(WG, IMLP, (p + 1) * KPANEL, n0, lane,
                          (p & 1) ? Pg0 : Pg1);
        issue_panel_async(WU, IMLP, (p + 1) * KPANEL, n0, lane,
                          (p & 1) ? Pu0 : Pu1);
        wait_async_le16();   // previous G+U pair (16 ops) fully landed
      } else {
        wait_async();        // last panel: drain everything
      }
      const int kp0 = p * KPANEL;
#pragma unroll 4
      for (int kk = 0; kk < KPANEL; kk += 4) {
        v2f a = *(const v2f*)(Xs + m * XS + kp0 + kk + kh); // ds_load_b64
        v2f bg, bu;                                         // conflict-free LDS
        bg.x = pg[(kk + kh) * 16 + nL];
        bg.y = pg[(kk + kh + 1) * 16 + nL];
        bu.x = pu[(kk + kh) * 16 + nL];
        bu.y = pu[(kk + kh + 1) * 16 + nL];
        cg = wmma_f32_k4(a, bg, cg);
        cu = wmma_f32_k4(a, bu, cu);
      }
    }
#pragma unroll
    for (int r = 0; r < 8; ++r) {
      float g = cg[r], u = cu[r];
      float h = (g / (1.f + __expf(-g))) * u;  // silu(g) * u
      Hs[(r + rowOff) * HS + n0 + nL] = h;
    }
  }
  __syncthreads();

  // ---- Phase B: down projection, weight, scatter ----
  for (int nt = wave; nt < DMODEL / 16; nt += 4) {
    const int n0 = nt * 16;
    v8f c = {};
    issue_panel_async(WD, DMODEL, 0, n0, lane, Pg0);
    for (int p = 0; p < IMLP / KPANEL; ++p) {
      float* pd = (p & 1) ? Pg1 : Pg0;
      if (p + 1 < IMLP / KPANEL) {
        issue_panel_async(WD, DMODEL, (p + 1) * KPANEL, n0, lane,
                          (p & 1) ? Pg0 : Pg1);
        wait_async_le8();    // previous single panel (8 ops) landed
      } else {
        wait_async();
      }
      const int kp0 = p * KPANEL;
#pragma unroll 4
      for (int kk = 0; kk < KPANEL; kk += 4) {
        v2f a = *(const v2f*)(Hs + m * HS + kp0 + kk + kh);
        v2f b;
        b.x = pd[(kk + kh) * 16 + nL];
        b.y = pd[(kk + kh + 1) * 16 + nL];
        c = wmma_f32_k4(a, b, c);
      }
    }
#pragma unroll
    for (int r = 0; r < 8; ++r) {
      const int row = r + rowOff;
      float val = c[r] * twts[row];
      float* dst = y + (size_t)tids[row] * DMODEL + n0 + nL;
      if (gathered) atomic_add_f32(dst, val);
      else          *dst = val;   // shared expert initializes y
    }
  }
}

// ---------------------------------------------------------------------------
extern "C" void kernel_launch(void* const* d_in, const int* in_sizes, int n_in,
                              void* d_out, int out_size, void* d_ws, size_t ws_size,
                              hipStream_t stream) {
  const float* x   = (const float*)d_in[0];  // [1,2048,1024]
  const float* gw  = (const float*)d_in[1];  // [64,1024]
  const float* wg  = (const float*)d_in[2];  // [64,1024,512]
  const float* wu  = (const float*)d_in[3];  // [64,1024,512]
  const float* wd  = (const float*)d_in[4];  // [64,512,1024]
  const float* swg = (const float*)d_in[5];  // [1,1024,512]
  const float* swu = (const float*)d_in[6];  // [1,1024,512]
  const float* swd = (const float*)d_in[7];  // [1,512,1024]

  // d_out: y [T,D] f32 | router_logits [T,E] f32 | topk ids [T,K] i32
  float* y          = (float*)d_out;
  float* logits_out = y + (size_t)T_TOK * DMODEL;
  int*   ids_out    = (int*)(logits_out + (size_t)T_TOK * NEXP);

  char* ws = (char*)d_ws;
  unsigned* counts   = (unsigned*)(ws + 0);      // 64 u32
  unsigned* offsets  = (unsigned*)(ws + 256);    // 64 u32
  int*      topk_id  = (int*)  (ws + 1024);
  float*    topk_w   = (float*)(ws + 1024 + 1 * T_TOK * TOPK * 4);
  int*   bucket_tid  = (int*)  (ws + 1024 + 2 * T_TOK * TOPK * 4);
  float* bucket_w    = (float*)(ws + 1024 + 3 * T_TOK * TOPK * 4);

  router_kernel<<<T_TOK, 64, 0, stream>>>(x, gw, logits_out, ids_out,
                                          topk_id, topk_w);
  count_kernel<<<1, 64, 0, stream>>>(topk_id, counts, offsets);
  scatter_kernel<<<1, 64, 0, stream>>>(topk_id, topk_w, offsets,
                                       bucket_tid, bucket_w);

  const size_t smem =
      (size_t)(16 * XS + 16 * HS + 4 * WAVE_PANEL_FLOATS) * sizeof(float);
  // Shared expert first: plain stores initialize every y element once.
  expert_kernel<<<T_TOK / 16, 128, smem, stream>>>(
      x, swg, swu, swd, y, nullptr, nullptr, nullptr, nullptr, 0);
  // Routed experts: gathered tiles, fp32 atomic accumulate into y.
  expert_kernel<<<NEXP * 128, 128, smem, stream>>>(
      x, wg, wu, wd, y, bucket_tid, bucket_w, counts, offsets, 1);
}